// TripletCenter40LossAllClass_61057255080333
// MI455X (gfx1250) — compile-verified
//
#include <hip/hip_runtime.h>

// Triplet-center loss, MI455X (gfx1250, wave32, WMMA).
// inputs [524288,40] f32, targets [524288] i32, centers [40,40] f32
// out[0] = loss, out[1] = prec.

#define N_ROWS   524288
#define DIM      40
#define NCLS     40
#define NT       48              // classes padded to 3 WMMA n-tiles
#define XSTRIDE  44              // LDS row stride (dwords): conflict-free + 16B aligned
#define GSTRIDE  49              // LDS row stride for dot-product tile (conflict-free b32)
#define NB       1024
#define ROWS_PER_BLOCK 128       // 8 waves * 16 rows
#define ITERS    (N_ROWS / (ROWS_PER_BLOCK * NB))   // = 4, exact

typedef __attribute__((ext_vector_type(2))) float v2f;
typedef __attribute__((ext_vector_type(4))) float v4f;
typedef __attribute__((ext_vector_type(8))) float v8f;

__global__ __launch_bounds__(256)
void tcl_main(const float* __restrict__ X, const int* __restrict__ T,
              const float* __restrict__ C, float* __restrict__ partials) {
  __shared__ float cLDS[NT * XSTRIDE];       // B operand (classes x dims), padded
  __shared__ float c2LDS[NT];                // per-class |c|^2 (pad = 1e30)
  __shared__ float xLDS[8 * 16 * XSTRIDE];   // per-wave X tiles
  __shared__ float gLDS[8 * 16 * GSTRIDE];   // per-wave dot products (16x48)
  __shared__ float redLDS[256];

  const int tid  = threadIdx.x;
  const int wave = tid >> 5;
  const int lane = tid & 31;
  const int l16  = lane & 15;
  const int hi   = lane >> 4;    // 0 = lanes 0-15, 1 = lanes 16-31

  // ---- stage centers into LDS (rows 40..47 zeroed) + per-class c2 ----
  for (int e = tid; e < NCLS * DIM; e += 256) {
    int r = e / DIM, c = e % DIM;
    cLDS[r * XSTRIDE + c] = C[e];
  }
  for (int e = tid; e < (NT - NCLS) * XSTRIDE; e += 256) {
    int r = NCLS + e / XSTRIDE, c = e % XSTRIDE;
    cLDS[r * XSTRIDE + c] = 0.0f;
  }
  if (tid < NT) {
    float s = 1e30f;
    if (tid < NCLS) {
      s = 0.0f;
      for (int k = 0; k < DIM; ++k) { float v = C[tid * DIM + k]; s += v * v; }
    }
    c2LDS[tid] = s;
  }

  float* xW = &xLDS[wave * 16 * XSTRIDE];
  float* gW = &gLDS[wave * 16 * GSTRIDE];

  float lossAcc = 0.0f, precAcc = 0.0f;

  for (int it = 0; it < ITERS; ++it) {
    const long long base =
        ((long long)it * NB + blockIdx.x) * ROWS_PER_BLOCK + (long long)wave * 16;
    __syncthreads();   // cLDS ready / previous iteration's LDS reads done

    // ---- coalesced load of 16x40 fp32 tile into LDS (160 float4 per wave) ----
    const float* src = X + base * DIM;
    if (it + 1 < ITERS) {
      // hint next tile toward the caches (global_prefetch_b8)
      __builtin_prefetch(src + (long long)NB * ROWS_PER_BLOCK * DIM, 0, 1);
    }
#pragma unroll
    for (int i = 0; i < 5; ++i) {
      int f = lane + 32 * i;                 // float4 index; 40%4==0 -> no row straddle
      int row = f / 10, col4 = (f % 10) * 4;
      v4f v = *(const v4f*)(src + f * 4);
      *(v4f*)&xW[row * XSTRIDE + col4] = v;
    }
    __syncthreads();

    // ---- G(16x48) = Xtile(16x40) x centers^T(40x48) via v_wmma_f32_16x16x4_f32 ----
    // A 16x4 f32 layout: lane m=lane&15; VGPR0=K(+0|+2 by half-wave), VGPR1=K+1|+3.
    // B 4x16 f32 layout mirrors A with N=lane&15.
    v8f acc0 = {}, acc1 = {}, acc2 = {};
#pragma unroll
    for (int kk = 0; kk < DIM / 4; ++kk) {
      int koff = kk * 4 + hi * 2;            // even dword -> 8B-aligned v2f loads
      v2f a  = *(const v2f*)&xW[l16 * XSTRIDE + koff];
      v2f b0 = *(const v2f*)&cLDS[(l16 +  0) * XSTRIDE + koff];
      v2f b1 = *(const v2f*)&cLDS[(l16 + 16) * XSTRIDE + koff];
      v2f b2 = *(const v2f*)&cLDS[(l16 + 32) * XSTRIDE + koff];
      acc0 = __builtin_amdgcn_wmma_f32_16x16x4_f32(false, a, false, b0, (short)0, acc0, false, false);
      acc1 = __builtin_amdgcn_wmma_f32_16x16x4_f32(false, a, false, b1, (short)0, acc1, false, false);
      acc2 = __builtin_amdgcn_wmma_f32_16x16x4_f32(false, a, false, b2, (short)0, acc2, false, false);
    }

    // ---- scatter C/D fragments (VGPR r holds M=r | M=r+8 per half-wave) ----
#pragma unroll
    for (int r = 0; r < 8; ++r) {
      int m = r + hi * 8;
      gW[m * GSTRIDE + l16 +  0] = acc0[r];
      gW[m * GSTRIDE + l16 + 16] = acc1[r];
      gW[m * GSTRIDE + l16 + 32] = acc2[r];
    }
    __syncthreads();

    // ---- per-row reduction: lanes 0-15 each own one of the wave's 16 rows ----
    if (hi == 0) {
      float x2 = 0.0f;
      for (int k = 0; k < DIM; ++k) { float v = xW[l16 * XSTRIDE + k]; x2 += v * v; }
      int tgt = T[base + l16];
      float ap = 0.0f, an = 1e30f;
      for (int j = 0; j < NCLS; ++j) {
        float d2 = x2 + c2LDS[j] - 2.0f * gW[l16 * GSTRIDE + j];
        float d  = sqrtf(fmaxf(d2, 1e-12f));
        if (j == tgt) ap = d;
        else          an = fminf(an, d);
      }
      lossAcc += fmaxf(ap - an /* + MARGIN(0) */, 0.0f);
      precAcc += (an > ap) ? 1.0f : 0.0f;
    }
  }

  // ---- deterministic fixed-order block reduction ----
  __syncthreads();
  redLDS[tid] = lossAcc;
  __syncthreads();
  for (int s = 128; s > 0; s >>= 1) {
    if (tid < s) redLDS[tid] += redLDS[tid + s];
    __syncthreads();
  }
  if (tid == 0) partials[2 * blockIdx.x] = redLDS[0];
  __syncthreads();
  redLDS[tid] = precAcc;
  __syncthreads();
  for (int s = 128; s > 0; s >>= 1) {
    if (tid < s) redLDS[tid] += redLDS[tid + s];
    __syncthreads();
  }
  if (tid == 0) partials[2 * blockIdx.x + 1] = redLDS[0];
}

__global__ void tcl_reduce(const float* __restrict__ partials, float* __restrict__ out) {
  if (threadIdx.x == 0 && blockIdx.x == 0) {
    float ls = 0.0f, ps = 0.0f;
    for (int i = 0; i < NB; ++i) { ls += partials[2 * i]; ps += partials[2 * i + 1]; }
    out[0] = ls / (float)N_ROWS;
    out[1] = ps / (float)N_ROWS;
  }
}

extern "C" void kernel_launch(void* const* d_in, const int* in_sizes, int n_in,
                              void* d_out, int out_size, void* d_ws, size_t ws_size,
                              hipStream_t stream) {
  const float* X = (const float*)d_in[0];   // inputs  [524288,40] f32
  const int*   T = (const int*)d_in[1];     // targets [524288]    i32
  const float* C = (const float*)d_in[2];   // centers [40,40]     f32
  float* out      = (float*)d_out;          // {loss, prec}
  float* partials = (float*)d_ws;           // 2*NB floats, fully rewritten each call

  tcl_main<<<NB, 256, 0, stream>>>(X, T, C, partials);
  tcl_reduce<<<1, 32, 0, stream>>>(partials, out);
}